// GNNModel_11467562680520
// MI455X (gfx1250) — compile-verified
//
#include <hip/hip_runtime.h>

// GCNConv forward for MI455X (gfx1250, wave32).
//   h = x @ W                       (fp32 WMMA 16x16x4, LDS-staged via async-to-LDS)
//   deg[dst]++, self-loops, dinv=rsqrt(deg)
//   out[dst] += dinv[src]*dinv[dst]*h[src]   (wave-per-edge, f32 atomics)
//   out = relu(out + b)

typedef __attribute__((ext_vector_type(2))) float v2f;
typedef __attribute__((ext_vector_type(8))) float v8f;

#define IN_DIM 128
#define OUT_DIM 128
#define ROWS_PER_BLOCK 256   // 16 row-tiles of 16
#define ROW_TILES 16

// ---- CDNA5 async copy helpers (ASYNCcnt path, CDNA5 ISA) ----

__device__ __forceinline__ void async_copy_b128(unsigned lds_addr,
                                                unsigned long long gbase,
                                                unsigned goff) {
  // GVS mode: mem = SADDR + VADDR(32b) ; VDST = LDS byte address
  asm volatile("global_load_async_to_lds_b128 %0, %1, %2"
               :: "v"(lds_addr), "v"(goff), "s"(gbase) : "memory");
}

__device__ __forceinline__ void wait_async0() {
  asm volatile("s_wait_asynccnt 0x0" ::: "memory");
}

// ---------------- degree kernels ----------------

__global__ void k_deg_init(float* __restrict__ deg, int n) {
  int i = blockIdx.x * blockDim.x + threadIdx.x;
  if (i < n) deg[i] = 1.0f;  // self-loop
}

__global__ void k_deg_accum(const int* __restrict__ dst, float* __restrict__ deg, int e) {
  int i = blockIdx.x * blockDim.x + threadIdx.x;
  if (i < e) atomicAdd(&deg[dst[i]], 1.0f);
}

__global__ void k_deg_rsqrt(float* __restrict__ deg, int n) {
  int i = blockIdx.x * blockDim.x + threadIdx.x;
  if (i < n) deg[i] = rsqrtf(deg[i]);
}

// ---------------- GEMM: h = x @ W via V_WMMA_F32_16X16X4_F32 ----------------
// Block = 256 threads = 8 waves; wave w owns col-tile [16w,16w+16).
// Block covers 256 rows as 16 row-tiles; W (64KB) staged to LDS once,
// x tiles (8KB) double-buffered with async-to-LDS overlap.
//
// Fragment layouts (ISA 7.12.2, 32-bit, wave32):
//   A (16x4):  lanes 0-15 -> row M=lane, v0=K0,v1=K1 ; lanes 16-31 -> K2,K3
//   B (4x16):  lanes 0-15 -> col N=lane, v0=K0,v1=K1 ; lanes 16-31 -> K2,K3
//   C/D:       VGPR r: lanes 0-15 (M=r, N=lane), lanes 16-31 (M=r+8, N=lane-16)

__global__ void k_gemm_wmma(const float* __restrict__ x,
                            const float* __restrict__ W,
                            float* __restrict__ h, int n) {
  __shared__ float Ws[IN_DIM * OUT_DIM];        // 64 KB
  __shared__ float Xs[2][16 * IN_DIM];          // 2 x 8 KB double buffer

  const int tid  = threadIdx.x;
  const int wave = tid >> 5;
  const int lane = tid & 31;
  const int half = lane >> 4;     // 0: K={0,1}, 1: K={2,3}
  const int l15  = lane & 15;
  const int block_row = blockIdx.x * ROWS_PER_BLOCK;

  const unsigned wsBase = (unsigned)(unsigned long long)&Ws[0];
  const unsigned xsBase = (unsigned)(unsigned long long)&Xs[0][0];
  const unsigned long long wG = (unsigned long long)W;
  const unsigned long long xG = (unsigned long long)x;

  // --- stage W: 64 KB, 16 x b128 per thread, fully coalesced ---
#pragma unroll
  for (int i = 0; i < 16; ++i) {
    const unsigned off = (unsigned)tid * 16u + (unsigned)i * 4096u;
    async_copy_b128(wsBase + off, wG, off);
  }

  // --- stage x tile t into buffer buf: 16 rows x 512B, 2 x b128 per thread ---
  auto stage_x = [&](int t, int buf) {
    int row = block_row + t * 16 + (tid >> 4);          // 16 threads per row
    if (row >= n) row = n - 1;                           // clamp (valid mem)
    const unsigned col  = (unsigned)(tid & 15) * 32u;
    const unsigned goff = (unsigned)row * (IN_DIM * 4u) + col;
    const unsigned loff = xsBase + (unsigned)buf * (16u * IN_DIM * 4u)
                        + (unsigned)(tid >> 4) * (IN_DIM * 4u) + col;
    async_copy_b128(loff,      xG, goff);
    async_copy_b128(loff + 16, xG, goff + 16);
  };

  stage_x(0, 0);
  wait_async0();        // W + x tile 0 resident
  __syncthreads();

  const int bcol = wave * 16 + l15;
  const float* wb = &Ws[(half * 2) * OUT_DIM + bcol];   // B feed (LDS)

  for (int t = 0; t < ROW_TILES; ++t) {
    if (t + 1 < ROW_TILES) stage_x(t + 1, (t + 1) & 1); // overlap next tile

    const float* xa = &Xs[t & 1][l15 * IN_DIM + half * 2]; // A feed (LDS)

    v8f acc = {};
#pragma unroll 8
    for (int k0 = 0; k0 < IN_DIM; k0 += 4) {
      v2f a = *(const v2f*)(xa + k0);                   // ds_load_b64
      v2f b;
      b.x = wb[k0 * OUT_DIM];                           // ds_load_b32 x2
      b.y = wb[k0 * OUT_DIM + OUT_DIM];
      acc = __builtin_amdgcn_wmma_f32_16x16x4_f32(
          false, a, false, b, (short)0, acc, false, false);
    }

    // store 16x16 tile; uniform guard keeps the fast path branch-free
    const int m0 = block_row + t * 16;
    float* hp = h + (size_t)(m0 + half * 8) * OUT_DIM + bcol;
    if (m0 + 16 <= n) {
#pragma unroll
      for (int r = 0; r < 8; ++r) hp[(size_t)r * OUT_DIM] = acc[r];
    } else {
#pragma unroll
      for (int r = 0; r < 8; ++r)
        if (m0 + half * 8 + r < n) hp[(size_t)r * OUT_DIM] = acc[r];
    }

    if (t + 1 < ROW_TILES) {
      wait_async0();      // next x tile landed
      __syncthreads();    // all waves done with buf before it is refilled
    }
  }
}

// ---------------- self-loop init: out = dinv^2 * h (float4) ----------------

__global__ void k_self_init(const float4* __restrict__ h4,
                            const float* __restrict__ dinv,
                            float4* __restrict__ out4, int total4) {
  int i = blockIdx.x * blockDim.x + threadIdx.x;
  if (i < total4) {
    const float d = dinv[i >> 5];        // 32 float4 per node row
    const float s = d * d;
    float4 v = h4[i];
    v.x *= s; v.y *= s; v.z *= s; v.w *= s;
    out4[i] = v;
  }
}

// ---------------- edge scatter: one wave per edge ----------------

__global__ void k_edges(const int* __restrict__ ei,
                        const float* __restrict__ h,
                        const float* __restrict__ dinv,
                        float* __restrict__ out, int e) {
  const int edge = blockIdx.x * 8 + (threadIdx.x >> 5);  // 8 waves / block
  if (edge >= e) return;                                  // uniform per wave
  const int lane = threadIdx.x & 31;

  const int s = ei[edge];
  const int d = ei[e + edge];
  const float nrm = dinv[s] * dinv[d];

  const float4 v = ((const float4*)(h + (size_t)s * OUT_DIM))[lane];
  float* op = out + (size_t)d * OUT_DIM + lane * 4;
  atomicAdd(op + 0, v.x * nrm);
  atomicAdd(op + 1, v.y * nrm);
  atomicAdd(op + 2, v.z * nrm);
  atomicAdd(op + 3, v.w * nrm);
}

// ---------------- finalize: out = relu(out + bias) (float4) ----------------

__global__ void k_finalize(float4* __restrict__ out4, const float4* __restrict__ b4,
                           int total4) {
  int i = blockIdx.x * blockDim.x + threadIdx.x;
  if (i < total4) {
    const float4 bb = b4[i & 31];
    float4 v = out4[i];
    v.x = fmaxf(v.x + bb.x, 0.0f);
    v.y = fmaxf(v.y + bb.y, 0.0f);
    v.z = fmaxf(v.z + bb.z, 0.0f);
    v.w = fmaxf(v.w + bb.w, 0.0f);
    out4[i] = v;
  }
}

// ---------------- launcher ----------------

extern "C" void kernel_launch(void* const* d_in, const int* in_sizes, int n_in,
                              void* d_out, int out_size, void* d_ws, size_t ws_size,
                              hipStream_t stream) {
  const float* x    = (const float*)d_in[0];
  const int*   ei   = (const int*)d_in[1];   // [2, E] flat: src then dst
  const float* W    = (const float*)d_in[2];
  const float* bias = (const float*)d_in[3];
  float*       out  = (float*)d_out;

  const int N = in_sizes[0] / IN_DIM;
  const int E = in_sizes[1] / 2;
  const int total4 = N * (OUT_DIM / 4);

  float* h    = (float*)d_ws;                 // N*128 floats
  float* dinv = h + (size_t)N * OUT_DIM;      // N floats

  k_deg_init <<<(N + 255) / 256, 256, 0, stream>>>(dinv, N);
  k_deg_accum<<<(E + 255) / 256, 256, 0, stream>>>(ei + E, dinv, E);
  k_deg_rsqrt<<<(N + 255) / 256, 256, 0, stream>>>(dinv, N);

  k_gemm_wmma<<<(N + ROWS_PER_BLOCK - 1) / ROWS_PER_BLOCK, 256, 0, stream>>>(x, W, h, N);

  k_self_init<<<(total4 + 255) / 256, 256, 0, stream>>>(
      (const float4*)h, dinv, (float4*)out, total4);

  k_edges<<<(E + 7) / 8, 256, 0, stream>>>(ei, h, dinv, out, E);

  k_finalize<<<(total4 + 255) / 256, 256, 0, stream>>>(
      (float4*)out, (const float4*)bias, total4);
}